// SSKernelSlow_37993280700534
// MI455X (gfx1250) — compile-verified
//
#include <hip/hip_runtime.h>
#include <hip/hip_bf16.h>
#include <math.h>

typedef float v2f __attribute__((ext_vector_type(2)));
typedef float v8f __attribute__((ext_vector_type(8)));

constexpr int SN    = 64;    // state dim (fixed by reference)
constexpr int LDP   = 65;    // padded stride for 64-wide LDS planes (bank-conflict-free)
constexpr int LDW   = 130;   // padded stride for 129-wide augmented solve matrix
constexpr int NT    = 256;   // 8 wave32 per workgroup
constexpr int CHMAX = 4;

// LDS layout (floats):
//   [    0,  8320)  solve W_re (64x130)   | gemm Q_re(0..4160) Q_im(4160..8320)
//   [ 8320, 16640)  solve W_im (64x130)   | gemm Y_re / Y_im
//   [16640, 24960)  P_re / P_im   (dA, later dA^(2^s))
//   [24960, 33280)  X_re / X_im   (krylov block, 64x64 in 64x65)
//   [33280, 33792)  C re/im  (CHMAX*64 each)
//   [33792, 33920)  GJ elimination factors re/im
//   [33920, 33922)  pivot reciprocal broadcast
constexpr int SMEMF = 33924;

__device__ __forceinline__ v8f wmma_f32(v2f a, v2f b, v8f c) {
  // D(16x16) += A(16x4) * B(4x16), fp32 — CDNA5 V_WMMA_F32_16X16X4_F32
  return __builtin_amdgcn_wmma_f32_16x16x4_f32(false, a, false, b, (short)0, c,
                                               false, false);
}

// Complex GEMM: Y(64x64) = P(64x64) * X(64x64), all LDS planes, stride LDP.
// 8 waves; wave w owns output tiles (mt = w>>1, nt = {2*(w&1), 2*(w&1)+1}).
// Cre += Are*Bre + (-Aim)*Bim ; Cim += Are*Bim + Aim*Bre  (4 WMMAs / k-step / tile)
__device__ __forceinline__ void cgemm64(float* __restrict__ Yre, float* __restrict__ Yim,
                                        const float* __restrict__ Pre, const float* __restrict__ Pim,
                                        const float* __restrict__ Xre, const float* __restrict__ Xim,
                                        int tid) {
  const int wave = tid >> 5;
  const int lane = tid & 31;
  const int hl   = lane >> 4;       // half-wave select (K pair / M+8)
  const int l16  = lane & 15;
  const int mt   = wave >> 1;
  const int nt0  = (wave & 1) << 1;
  const int arow = mt * 16 + l16;   // A fragment: lane = M row
  const int c0   = nt0 * 16 + l16;  // B/C fragment: lane = N col
  const int c1   = c0 + 16;

  v8f ar0 = {}, ai0 = {}, ar1 = {}, ai1 = {};
#pragma unroll 4
  for (int k0 = 0; k0 < SN; k0 += 4) {
    const int ka = k0 + 2 * hl;     // lanes 0-15: K0/K1, lanes 16-31: K2/K3
    v2f a_re, a_im, a_imn, b_re0, b_im0, b_re1, b_im1;
    a_re.x = Pre[arow * LDP + ka];     a_re.y = Pre[arow * LDP + ka + 1];
    a_im.x = Pim[arow * LDP + ka];     a_im.y = Pim[arow * LDP + ka + 1];
    a_imn.x = -a_im.x;                 a_imn.y = -a_im.y;
    b_re0.x = Xre[ka * LDP + c0];      b_re0.y = Xre[(ka + 1) * LDP + c0];
    b_im0.x = Xim[ka * LDP + c0];      b_im0.y = Xim[(ka + 1) * LDP + c0];
    b_re1.x = Xre[ka * LDP + c1];      b_re1.y = Xre[(ka + 1) * LDP + c1];
    b_im1.x = Xim[ka * LDP + c1];      b_im1.y = Xim[(ka + 1) * LDP + c1];
    ar0 = wmma_f32(a_re,  b_re0, ar0);
    ar0 = wmma_f32(a_imn, b_im0, ar0);
    ai0 = wmma_f32(a_re,  b_im0, ai0);
    ai0 = wmma_f32(a_im,  b_re0, ai0);
    ar1 = wmma_f32(a_re,  b_re1, ar1);
    ar1 = wmma_f32(a_imn, b_im1, ar1);
    ai1 = wmma_f32(a_re,  b_im1, ai1);
    ai1 = wmma_f32(a_im,  b_re1, ai1);
  }
#pragma unroll
  for (int i = 0; i < 8; ++i) {       // C/D: vgpr i -> row i (lanes<16) / i+8
    const int row = mt * 16 + 8 * hl + i;
    Yre[row * LDP + c0] = ar0[i];
    Yim[row * LDP + c0] = ai0[i];
    Yre[row * LDP + c1] = ar1[i];
    Yim[row * LDP + c1] = ai1[i];
  }
}

extern "C" __global__ __launch_bounds__(NT)
void s4_krylov_fused(const float* __restrict__ A, const float* __restrict__ B,
                     const float* __restrict__ C, const float* __restrict__ log_dt,
                     const int* __restrict__ Lp, float* __restrict__ out,
                     int H, int CH) {
  __shared__ float smem[SMEMF];
  float* Wre = smem;             float* Wim = smem + 8320;
  float* Qre = smem;             float* Qim = smem + 4160;
  float* Yre = smem + 8320;      float* Yim = smem + 12480;
  float* Pre = smem + 16640;     float* Pim = smem + 20800;
  float* Xre = smem + 24960;     float* Xim = smem + 29120;
  float* Cre = smem + 33280;     float* Cim = smem + 33280 + CHMAX * SN;
  float* facre = smem + 33792;   float* facim = smem + 33856;
  float* piv = smem + 33920;

  const int h   = blockIdx.x;
  const int tid = threadIdx.x;
  const int L   = *Lp;
  const float dt = __expf(log_dt[h]);
  const float hd = 0.5f * dt;

  // ---- Phase 1: augmented system [ I - hd*A | I + hd*A | dt*B ] ----
  const float* Ah = A + (size_t)h * SN * SN * 2;
  for (int idx = tid; idx < SN * SN; idx += NT) {
    const int r = idx >> 6, c = idx & 63;
    const float are = Ah[2 * idx], aim = Ah[2 * idx + 1];
    const float dg = (r == c) ? 1.0f : 0.0f;
    Wre[r * LDW + c]      = dg - hd * are;
    Wim[r * LDW + c]      =    - hd * aim;
    Wre[r * LDW + 64 + c] = dg + hd * are;
    Wim[r * LDW + 64 + c] =      hd * aim;
  }
  if (tid < SN) {
    Wre[tid * LDW + 128] = dt * B[(h * SN + tid) * 2];
    Wim[tid * LDW + 128] = dt * B[(h * SN + tid) * 2 + 1];
  }
  for (int idx = tid; idx < CH * SN; idx += NT) {
    const int ch = idx >> 6, n = idx & 63;
    Cre[idx] = C[((size_t)(ch * H + h) * SN + n) * 2];
    Cim[idx] = C[((size_t)(ch * H + h) * SN + n) * 2 + 1];
  }
  __syncthreads();

  // ---- Phase 2: Gauss-Jordan, no pivoting (A_back Hermitian part = (1+dt/4)I > 0) ----
  for (int k = 0; k < SN; ++k) {
    if (tid == 0) {
      const float pr = Wre[k * LDW + k], pi = Wim[k * LDW + k];
      const float inv = 1.0f / (pr * pr + pi * pi);
      piv[0] = pr * inv;  piv[1] = -pi * inv;
    }
    __syncthreads();
    const float ivr = piv[0], ivi = piv[1];
    if (tid < 129) {                       // scale pivot row
      const float wr = Wre[k * LDW + tid], wi = Wim[k * LDW + tid];
      Wre[k * LDW + tid] = wr * ivr - wi * ivi;
      Wim[k * LDW + tid] = wr * ivi + wi * ivr;
    } else if (tid >= 136 && tid < 200) {  // record elimination factors
      const int r = tid - 136;
      facre[r] = (r == k) ? 0.0f : Wre[r * LDW + k];
      facim[r] = (r == k) ? 0.0f : Wim[r * LDW + k];
    }
    __syncthreads();
    for (int idx = tid; idx < SN * 129; idx += NT) {
      const int r = idx / 129, c = idx - r * 129;
      if (r == k) continue;
      const float fr = facre[r], fi = facim[r];
      const float wr = Wre[k * LDW + c], wi = Wim[k * LDW + c];
      Wre[r * LDW + c] -= fr * wr - fi * wi;
      Wim[r * LDW + c] -= fr * wi + fi * wr;
    }
    __syncthreads();
  }

  // ---- Phase 3: dA -> P, dB -> X[:,0] (distinct LDS regions from W) ----
  for (int idx = tid; idx < SN * SN; idx += NT) {
    const int r = idx >> 6, c = idx & 63;
    Pre[r * LDP + c] = Wre[r * LDW + 64 + c];
    Pim[r * LDP + c] = Wim[r * LDW + 64 + c];
  }
  if (tid < SN) {
    Xre[tid * LDP] = Wre[tid * LDW + 128];
    Xim[tid * LDP] = Wim[tid * LDW + 128];
  }
  __syncthreads();

  // ---- Phase 4: doubling — X0 = [dB, dA dB, ..., dA^63 dB], P -> dA^64 ----
  float *pre = Pre, *pim = Pim, *qre = Qre, *qim = Qim;
  for (int s = 0; s < 6; ++s) {
    const int w = 1 << s;
    for (int idx = tid; idx < SN * w; idx += NT) {   // X[:, w+l] = P * X[:, l]
      const int l = idx >> 6, n = idx & 63;
      float sre = 0.f, sim = 0.f;
      for (int m = 0; m < SN; ++m) {
        const float pr = pre[n * LDP + m], pq = pim[n * LDP + m];
        const float xr = Xre[m * LDP + l], xi = Xim[m * LDP + l];
        sre += pr * xr - pq * xi;
        sim += pr * xi + pq * xr;
      }
      Xre[n * LDP + w + l] = sre;
      Xim[n * LDP + w + l] = sim;
    }
    cgemm64(qre, qim, pre, pim, pre, pim, tid);      // Q = P*P (WMMA)
    __syncthreads();
    float* t;
    t = pre; pre = qre; qre = t;
    t = pim; pim = qim; qim = t;
  }

  // ---- Phase 5: emit Re(C . X_j), X_{j+1} = dA^64 * X_j (WMMA) ----
  float *xre = Xre, *xim = Xim, *yre = Yre, *yim = Yim;
  const int nchunks = (L + 63) >> 6;
  for (int j = 0; j < nchunks; ++j) {
    if (tid < SN) {
      const int l = tid, lg = (j << 6) + l;
      if (lg < L) {
        for (int ch = 0; ch < CH; ++ch) {
          float acc = 0.f;
          for (int n = 0; n < SN; ++n)
            acc += Cre[ch * SN + n] * xre[n * LDP + l]
                 - Cim[ch * SN + n] * xim[n * LDP + l];
          out[((size_t)ch * H + h) * L + lg] = acc;
        }
      }
    }
    if (j + 1 < nchunks) {
      cgemm64(yre, yim, pre, pim, xre, xim, tid);
      __syncthreads();
      float* t;
      t = xre; xre = yre; yre = t;
      t = xim; xim = yim; yim = t;
    }
  }
}

extern "C" void kernel_launch(void* const* d_in, const int* in_sizes, int n_in,
                              void* d_out, int out_size, void* d_ws, size_t ws_size,
                              hipStream_t stream) {
  const float* A      = (const float*)d_in[0];   // (H,N,N,2)
  const float* B      = (const float*)d_in[1];   // (H,N,2)
  const float* C      = (const float*)d_in[2];   // (CH,H,N,2)
  const float* log_dt = (const float*)d_in[3];   // (H,1)
  const int*   Lp     = (const int*)d_in[4];     // scalar L
  const int H  = in_sizes[3];                    // log_dt flat count == H
  int CH = in_sizes[2] / (H * SN * 2);
  if (CH < 1) CH = 1;
  if (CH > CHMAX) CH = CHMAX;
  s4_krylov_fused<<<dim3(H), dim3(NT), 0, stream>>>(A, B, C, log_dt, Lp,
                                                    (float*)d_out, H, CH);
}